// SelfAttention_61710090109602
// MI455X (gfx1250) — compile-verified
//
#include <hip/hip_runtime.h>
#include <math.h>

// Problem constants (from reference): B=32, S=4096, D=512, fp32.
#define B_      32
#define S_      4096
#define D_      512
#define CHUNKS  16           // S-splits per batch -> 32*16 = 512 workgroups
#define SC      (S_/CHUNKS)  // 256 rows per workgroup
#define T_      32           // tile rows staged in LDS (32*512*4 = 64 KB)
#define NTILES  (SC/T_)      // 8 tiles per workgroup

typedef __attribute__((ext_vector_type(2))) float v2f;
typedef __attribute__((ext_vector_type(8))) float v8f;
typedef __attribute__((ext_vector_type(4))) unsigned int u32x4;
typedef __attribute__((ext_vector_type(8))) int i32x8;
typedef __attribute__((ext_vector_type(4))) int i32x4;

// Issue one TDM transfer: 2D tile (512 cols x 32 rows, 4B elements) from
// global memory into LDS at byte offset lds_off.  D# layout per ISA ch.8.
static __device__ __forceinline__ void tdm_load_tile(const float* gsrc,
                                                     unsigned lds_off) {
    const unsigned long long ga = (unsigned long long)(uintptr_t)gsrc;
    u32x4 g0;
    g0.x = 1u;                                   // count=1 (valid user D#)
    g0.y = lds_off;                              // lds_addr (bytes)
    g0.z = (unsigned)(ga & 0xFFFFFFFFu);         // global_addr[31:0]
    g0.w = (unsigned)((ga >> 32) & 0x01FFFFFFu)  // global_addr[56:32]
         | (2u << 30);                           // type=2 ("image")
    i32x8 g1;
    g1[0] = (int)(2u << 16);        // workgroup_mask=0, data_size=2 (4 bytes)
    g1[1] = (int)(512u << 16);      // tensor_dim0 = 512 (bits[63:48])
    g1[2] = (int)(32u  << 16);      // tensor_dim0 hi=0 | tensor_dim1 = 32
    g1[3] = (int)(512u << 16);      // tensor_dim1 hi=0 | tile_dim0 = 512
    g1[4] = 32;                     // tile_dim1 = 32, tile_dim2 = 0
    g1[5] = 512;                    // tensor_dim0_stride = 512 elements
    g1[6] = 0;                      // stride hi / dim1_stride lo (unused, 2D)
    g1[7] = 0;
    const i32x4 z4 = {0, 0, 0, 0};  // groups 2/3: unused for 2D tensors
#if defined(__clang_major__) && (__clang_major__ >= 23)
    const i32x8 z8 = {0, 0, 0, 0, 0, 0, 0, 0};
    __builtin_amdgcn_tensor_load_to_lds(g0, g1, z4, z4, z8, 0);
#else
    __builtin_amdgcn_tensor_load_to_lds(g0, g1, z4, z4, 0);
#endif
}

// Kernel 1: per (batch, S-chunk) -> online-softmax partials (m, l, acc[D]).
// Block = 512 threads = 16 wave32; thread tid owns output column d = tid.
__global__ __launch_bounds__(512)
void attn_pool_partial(const float* __restrict__ ctx,
                       const unsigned char* __restrict__ mask,
                       const float* __restrict__ w,
                       const float* __restrict__ bias_p,
                       float* __restrict__ part_acc,   // [B*CHUNKS][D]
                       float* __restrict__ part_ml)    // [B*CHUNKS][2]
{
    __shared__ float tile[2][T_ * D_];  // 2 x 64 KB TDM double buffer
    __shared__ float sw[D_];            // w vector, 2 KB
    __shared__ float probs[T_];         // per-tile scores -> probabilities
    __shared__ float sred[2];           // tile max / tile sum broadcast

    const int tid  = threadIdx.x;
    const int lane = tid & 31;
    const int wave = tid >> 5;
    const int b    = blockIdx.x;
    const int chunk= blockIdx.y;
    const int s0   = chunk * SC;
    const float bias = bias_p[0];

    sw[tid] = w[tid];                   // D_ == blockDim, one element each

    float m   = -INFINITY;              // running max   (uniform across block)
    float l   = 0.f;                    // running denom (uniform across block)
    float acc = 0.f;                    // running numerator for column d = tid

    const float* cbase = ctx + ((size_t)b * S_ + s0) * (size_t)D_;
    const unsigned char* mbase = mask + (size_t)b * S_ + s0;
    const unsigned lds_buf0 = (unsigned)(uintptr_t)&tile[0][0];
    const unsigned lds_buf1 = (unsigned)(uintptr_t)&tile[1][0];

    // Prologue: TDM-prefetch tile 0 into buffer 0 (wave 0 issues, EXEC full).
    if (wave == 0) tdm_load_tile(cbase, lds_buf0);

    for (int t = 0; t < NTILES; ++t) {
        const int cur = t & 1;
        __syncthreads();                // pooling of tile t-1 done -> buf free
        if (wave == 0) {
            if (t + 1 < NTILES) {       // pipeline: stage tile t+1 now
                tdm_load_tile(cbase + (size_t)(t + 1) * T_ * D_,
                              cur ? lds_buf0 : lds_buf1);
                __builtin_amdgcn_s_wait_tensorcnt(1);  // tile t landed
            } else {
                __builtin_amdgcn_s_wait_tensorcnt(0);
            }
        }
        __syncthreads();                // tile[cur] visible to all waves
        const float* tl = &tile[cur][0];

        // ---- scores via V_WMMA_F32_16X16X4_F32: waves 0,1 take 16 rows each.
        // A (16x4 f32): lanes 0-15 hold K=0,1 in v0,v1; lanes 16-31 hold K=2,3.
        // B (4x16) = w replicated across all 16 N columns -> per-lane B pair
        // follows the same K selection as A. All D columns equal the score.
        if (wave < 2) {
            const int hi  = lane >> 4;          // 0: K=0,1  1: K=2,3
            const int row = wave * 16 + (lane & 15);
            v8f c = {};
            #pragma unroll 4
            for (int ks = 0; ks < D_ / 4; ++ks) {
                const int kb = ks * 4 + hi * 2;
                v2f a;  a.x  = tl[row * D_ + kb];  a.y  = tl[row * D_ + kb + 1];
                v2f bb; bb.x = sw[kb];             bb.y = sw[kb + 1];
                c = __builtin_amdgcn_wmma_f32_16x16x4_f32(
                        /*neg_a=*/false, a, /*neg_b=*/false, bb,
                        /*c_mod=*/(short)0, c, /*reuse_a=*/false, /*reuse_b=*/false);
            }
            // C/D layout: VGPR g -> M=g (lanes 0-15) / M=8+g (lanes 16-31).
            if (lane == 0 || lane == 16) {
                const int rbase = wave * 16 + hi * 8;
                #pragma unroll
                for (int g = 0; g < 8; ++g) {
                    const int r  = rbase + g;
                    const int sg = t * T_ + r;          // row within chunk
                    float sc = c[g] + bias;
                    if (mbase[sg]) sc = -INFINITY;      // masked_fill(mask,-inf)
                    probs[r] = sc;
                }
            }
        }
        __syncthreads();

        // ---- tile max ----
        if (tid == 0) {
            float mt = -INFINITY;
            for (int i = 0; i < T_; ++i) mt = fmaxf(mt, probs[i]);
            sred[0] = mt;
        }
        __syncthreads();
        const float m_new = fmaxf(m, sred[0]);
        const float scale = (m_new == -INFINITY) ? 1.f : __expf(m - m_new);

        // ---- exponentiate tile scores ----
        if (tid < T_) {
            const float sc = probs[tid];
            probs[tid] = (sc == -INFINITY) ? 0.f : __expf(sc - m_new);
        }
        __syncthreads();
        if (tid == 0) {
            float su = 0.f;
            for (int i = 0; i < T_; ++i) su += probs[i];
            sred[1] = su;
        }
        __syncthreads();

        // ---- online update + pooling: acc_d += sum_s p_s * tile[s][d] ----
        l   = l * scale + sred[1];
        acc = acc * scale;
        #pragma unroll 8
        for (int i = 0; i < T_; ++i)
            acc = fmaf(probs[i], tl[i * D_ + tid], acc);
        m = m_new;
    }

    const int g = b * CHUNKS + chunk;
    part_acc[(size_t)g * D_ + tid] = acc;
    if (tid == 0) { part_ml[g * 2 + 0] = m; part_ml[g * 2 + 1] = l; }
}

// Kernel 2: merge the CHUNKS partials per batch (flash-decoding combine).
__global__ __launch_bounds__(512)
void attn_pool_merge(const float* __restrict__ part_acc,
                     const float* __restrict__ part_ml,
                     float* __restrict__ out)
{
    const int b = blockIdx.x;
    const int d = threadIdx.x;

    float M = -INFINITY;
    #pragma unroll
    for (int c = 0; c < CHUNKS; ++c)
        M = fmaxf(M, part_ml[(b * CHUNKS + c) * 2 + 0]);

    float L = 0.f, o = 0.f;
    #pragma unroll
    for (int c = 0; c < CHUNKS; ++c) {
        const int g = b * CHUNKS + c;
        const float coef = __expf(part_ml[g * 2 + 0] - M);
        L += coef * part_ml[g * 2 + 1];
        o  = fmaf(coef, part_acc[(size_t)g * D_ + d], o);
    }
    out[(size_t)b * D_ + d] = o / L;
}

extern "C" void kernel_launch(void* const* d_in, const int* in_sizes, int n_in,
                              void* d_out, int out_size, void* d_ws, size_t ws_size,
                              hipStream_t stream) {
    const float*         ctx  = (const float*)d_in[0];          // [B,S,D] f32
    const unsigned char* mask = (const unsigned char*)d_in[1];  // [B,S] bool (1B)
    const float*         w    = (const float*)d_in[2];          // [D] f32
    const float*         bias = (const float*)d_in[3];          // scalar f32

    float* part_acc = (float*)d_ws;                              // B*CHUNKS*D floats
    float* part_ml  = part_acc + (size_t)B_ * CHUNKS * D_;       // B*CHUNKS*2 floats

    dim3 grid1(B_, CHUNKS);
    attn_pool_partial<<<grid1, 512, 0, stream>>>(ctx, mask, w, bias, part_acc, part_ml);
    attn_pool_merge<<<B_, D_, 0, stream>>>(part_acc, part_ml, (float*)d_out);
}